// GNN_22857815949796
// MI455X (gfx1250) — compile-verified
//
#include <hip/hip_runtime.h>

typedef __attribute__((ext_vector_type(2))) float v2f;
typedef __attribute__((ext_vector_type(8))) float v8f;

// ---------------------------------------------------------------------------
// Degree / normalization kernels.
// edge_index is int64 (little-endian, values < 2^31) -> read low 32-bit word.
// ---------------------------------------------------------------------------
__global__ void gcn_init_deg(float* __restrict__ deg, int n) {
    int i = blockIdx.x * blockDim.x + threadIdx.x;
    if (i < n) deg[i] = 1.0f;  // self-loop contribution
}

__global__ void gcn_accum_deg(const int* __restrict__ dst32,  // int64 viewed as int32 pairs
                              float* __restrict__ deg, int E) {
    int e = blockIdx.x * blockDim.x + threadIdx.x;
    if (e < E) atomicAdd(deg + dst32[2 * e], 1.0f);
}

__global__ void gcn_rsqrt_deg(float* __restrict__ deg, int n) {
    int i = blockIdx.x * blockDim.x + threadIdx.x;
    if (i < n) deg[i] = rsqrtf(deg[i]);  // deg >= 1 always (self loops)
}

// ---------------------------------------------------------------------------
// Fused feature transform: H = act(X) @ W via V_WMMA_F32_16X16X4_F32.
// One wave32 per 16-row tile; K=16 = 4 chained 16x16x4 WMMAs (fp32 exact).
// Also seeds AGG with the self-loop term H[r]*dis[r]^2.
// Full tiles (the only case for N=100k) take a straight-line unguarded
// epilogue; the guarded path exists only for generality.
// ---------------------------------------------------------------------------
template <bool RELU_BIAS>
__global__ void gcn_gemm_wmma(const float* __restrict__ X,    // [n,16]
                              const float* __restrict__ W,    // [16,16] row-major
                              const float* __restrict__ bias, // [16] (RELU_BIAS only)
                              const float* __restrict__ dis,  // [n]
                              float* __restrict__ H,          // [n,16] out
                              float* __restrict__ AGG,        // [n,16] out
                              int n) {
    const int wave  = (int)((blockIdx.x * blockDim.x + threadIdx.x) >> 5);
    const int lane  = threadIdx.x & 31;
    const int tile0 = wave * 16;
    if (tile0 >= n) return;  // wave-uniform; EXEC stays all-1s otherwise

    const int half = lane >> 4;   // 0: K {k0,k0+1}, 1: K {k0+2,k0+3}
    const int m    = lane & 15;   // A row / B col / D col
    int rowA = tile0 + m;
    rowA = rowA < (n - 1) ? rowA : (n - 1);  // select, no EXEC change

    v8f c = {};
#pragma unroll
    for (int k0 = 0; k0 < 16; k0 += 4) {
        const int kk = k0 + half * 2;
        v2f a;
        {
            float x0 = X[rowA * 16 + kk + 0];
            float x1 = X[rowA * 16 + kk + 1];
            if (RELU_BIAS) {
                x0 += bias[kk + 0];
                x1 += bias[kk + 1];
                x0 = x0 > 0.0f ? x0 : 0.0f;
                x1 = x1 > 0.0f ? x1 : 0.0f;
            }
            a.x = x0; a.y = x1;
        }
        v2f b;
        b.x = W[(kk + 0) * 16 + m];
        b.y = W[(kk + 1) * 16 + m];
        c = __builtin_amdgcn_wmma_f32_16x16x4_f32(
                /*neg_a=*/false, a, /*neg_b=*/false, b,
                /*c_mod=*/(short)0, c, /*reuse_a=*/false, /*reuse_b=*/false);
    }

    // C/D layout: VGPR v -> row (8*half + v), col = lane&15
    const int col   = m;
    const int rbase = tile0 + half * 8;
    if (tile0 + 16 <= n) {
        // Full tile: unguarded straight-line stores (always taken for N%16==0)
#pragma unroll
        for (int v = 0; v < 8; ++v) {
            const int r  = rbase + v;
            const float hv = c[v];
            const float ds = dis[r];
            H[r * 16 + col]   = hv;
            AGG[r * 16 + col] = hv * ds * ds;
        }
    } else {
#pragma unroll
        for (int v = 0; v < 8; ++v) {
            const int r = rbase + v;
            if (r < n) {
                const float hv = c[v];
                const float ds = dis[r];
                H[r * 16 + col]   = hv;
                AGG[r * 16 + col] = hv * ds * ds;
            }
        }
    }
}

// ---------------------------------------------------------------------------
// Edge scatter: 16 lanes per edge -> 64B contiguous gather of H[src] and a
// 64B contiguous burst of global_atomic_add_f32 into AGG[dst]. The full
// working set (~50 MB with int32 index reads) is L2-resident on MI455X.
// ---------------------------------------------------------------------------
__global__ void gcn_edge_pass(const int* __restrict__ src32,  // int64 viewed as int32 pairs
                              const int* __restrict__ dst32,
                              const float* __restrict__ dis,
                              const float* __restrict__ H,
                              float* __restrict__ AGG, int E) {
    const long long gid = (long long)blockIdx.x * blockDim.x + threadIdx.x;
    const int e = (int)(gid >> 4);
    const int c = (int)(gid & 15);
    if (e >= E) return;
    const int s = src32[2 * e];
    const int d = dst32[2 * e];
    const float coef = dis[s] * dis[d];
    atomicAdd(AGG + d * 16 + c, H[s * 16 + c] * coef);
}

// ---------------------------------------------------------------------------
// Head: out[i] = relu(AGG2[i,:] + b2) . Wfc + bfc
// ---------------------------------------------------------------------------
__global__ void gcn_head(const float* __restrict__ AGG,
                         const float* __restrict__ b2,
                         const float* __restrict__ Wfc,
                         const float* __restrict__ bfc,
                         float* __restrict__ out, int n) {
    const int i = blockIdx.x * blockDim.x + threadIdx.x;
    if (i >= n) return;
    float acc = bfc[0];
#pragma unroll
    for (int c = 0; c < 16; ++c) {
        float h = AGG[i * 16 + c] + b2[c];
        h = h > 0.0f ? h : 0.0f;
        acc += h * Wfc[c];
    }
    out[i] = acc;
}

// ---------------------------------------------------------------------------
// Launch
// ---------------------------------------------------------------------------
extern "C" void kernel_launch(void* const* d_in, const int* in_sizes, int n_in,
                              void* d_out, int out_size, void* d_ws, size_t ws_size,
                              hipStream_t stream) {
    const float* x   = (const float*)d_in[0];
    const int*   ei  = (const int*)d_in[1];   // int64 data viewed as int32 pairs
    const float* W1  = (const float*)d_in[2];
    const float* b1  = (const float*)d_in[3];
    const float* W2  = (const float*)d_in[4];
    const float* b2  = (const float*)d_in[5];
    const float* Wfc = (const float*)d_in[6];
    const float* bfc = (const float*)d_in[7];
    float*       out = (float*)d_out;

    const int n = in_sizes[0] / 16;
    const int E = in_sizes[1] / 2;
    const int* src32 = ei;            // edge_index[0], int64 elements
    const int* dst32 = ei + 2LL * E;  // edge_index[1], int64 elements

    // Workspace layout: dis [n] | H [n*16] | AGG [n*16]   (~13.2 MB)
    float* ws = (float*)d_ws;
    const size_t nAl = ((size_t)n + 63) & ~(size_t)63;
    float* dis = ws;
    float* H   = ws + nAl;
    float* AGG = H + nAl * 16;

    const int T  = 256;
    const int nb = (n + T - 1) / T;
    const int eb = (E + T - 1) / T;
    const long long etot = (long long)E * 16;
    const int eb16 = (int)((etot + T - 1) / T);

    // 1) degrees (self-loop folded into init) -> dis = rsqrt(deg)
    gcn_init_deg<<<nb, T, 0, stream>>>(dis, n);
    gcn_accum_deg<<<eb, T, 0, stream>>>(dst32, dis, E);
    gcn_rsqrt_deg<<<nb, T, 0, stream>>>(dis, n);

    // GEMM grid: one wave32 per 16-row tile, 4 waves / 128-thread block
    const int tiles = (n + 15) / 16;
    const int gblk  = (tiles + 3) / 4;

    // 2) layer 1: H = x@W1 ; AGG = self-loop init
    gcn_gemm_wmma<false><<<gblk, 128, 0, stream>>>(x, W1, nullptr, dis, H, AGG, n);
    // 3) layer 1 scatter
    gcn_edge_pass<<<eb16, T, 0, stream>>>(src32, dst32, dis, H, AGG, E);
    // 4) layer 2: fuse (+b1, relu) into GEMM input; H = relu(AGG+b1)@W2
    gcn_gemm_wmma<true><<<gblk, 128, 0, stream>>>(AGG, W2, b1, dis, H, AGG, n);
    // 5) layer 2 scatter
    gcn_edge_pass<<<eb16, T, 0, stream>>>(src32, dst32, dis, H, AGG, E);
    // 6) head: relu(+b2) and 16->1 matvec fused
    gcn_head<<<nb, T, 0, stream>>>(AGG, b2, Wfc, bfc, out, n);
}